// MultiHeadAttentionLayer_25116968747116
// MI455X (gfx1250) — compile-verified
//
#include <hip/hip_runtime.h>
#include <hip/hip_bf16.h>

typedef __attribute__((ext_vector_type(2))) float v2f;
typedef __attribute__((ext_vector_type(8))) float v8f;

#define IN_DIM  128
#define HD      128
#define E_DIM   32
#define NHEAD   4
#define DHEAD   32
#define OUT_DIM 32
#define INV_SQRT_D 0.17677669529663687f  // 1/sqrt(32)

static __device__ __forceinline__ v8f wmma_f32(v2f a, v2f b, v8f c) {
  // D = A(16x4,f32) * B(4x16,f32) + C(16x16,f32); emits v_wmma_f32_16x16x4_f32
  return __builtin_amdgcn_wmma_f32_16x16x4_f32(false, a, false, b, (short)0, c,
                                               false, false);
}

// ---------------------------------------------------------------------------
// Zero the scatter accumulators (sums[N*32] ++ cnt[N], contiguous)
// ---------------------------------------------------------------------------
__global__ void zero_kernel(float* p, int n) {
  int i = blockIdx.x * blockDim.x + threadIdx.x;
  if (i < n) p[i] = 0.0f;
}

// ---------------------------------------------------------------------------
// Fold Wo into Wv:  Wvo[k, h*32+j] = sum_d Wv[k, h*32+d] * Wo[(h*32+d)*32 + j]
//                   bvo[h*32+j]    = sum_d bv[h*32+d]    * Wo[(h*32+d)*32 + j]
// ---------------------------------------------------------------------------
__global__ void precompute_wvo(const float* __restrict__ Wv,
                               const float* __restrict__ bv,
                               const float* __restrict__ Wo,
                               float* __restrict__ Wvo,
                               float* __restrict__ bvo) {
  int idx = blockIdx.x * blockDim.x + threadIdx.x;
  if (idx < IN_DIM * HD) {
    int k = idx >> 7;          // row of Wv (input dim)
    int c = idx & 127;         // output column h*32+j
    int h = c >> 5, j = c & 31;
    float acc = 0.0f;
    for (int d = 0; d < DHEAD; ++d)
      acc += Wv[k * HD + h * DHEAD + d] * Wo[(h * DHEAD + d) * OUT_DIM + j];
    Wvo[idx] = acc;
  } else if (idx < IN_DIM * HD + HD) {
    int c = idx - IN_DIM * HD;
    int h = c >> 5, j = c & 31;
    float acc = 0.0f;
    for (int d = 0; d < DHEAD; ++d)
      acc += bv[h * DHEAD + d] * Wo[(h * DHEAD + d) * OUT_DIM + j];
    bvo[c] = acc;
  }
}

// ---------------------------------------------------------------------------
// Fused node GEMM: Q = x@Wq+bq ; K = x@Wk+bk ; Vo = x@Wvo+bvo  (all [N,128])
// 256 threads = 8 waves; block owns a 16-row tile of x (LDS), wave w owns the
// 16-column tile n0 = 16*w of all three outputs. 32 k-steps of f32 WMMA.
// ---------------------------------------------------------------------------
__global__ __launch_bounds__(256) void node_gemm(
    const float* __restrict__ x,
    const float* __restrict__ Wq, const float* __restrict__ bq,
    const float* __restrict__ Wk, const float* __restrict__ bk,
    const float* __restrict__ Wvo, const float* __restrict__ bvo,
    float* __restrict__ Q, float* __restrict__ Km, float* __restrict__ Vo,
    int N) {
  __shared__ float As[16][IN_DIM];
  const int tid = threadIdx.x;
  const int m0 = blockIdx.x * 16;

  // cooperative A-tile load (2048 floats / 256 threads)
  for (int i = 0; i < 8; ++i) {
    int lin = tid + i * 256;
    int r = lin >> 7, c = lin & 127;
    int gr = m0 + r; if (gr >= N) gr = N - 1;
    As[r][c] = x[(size_t)gr * IN_DIM + c];
  }
  __syncthreads();

  const int w = tid >> 5, L = tid & 31;
  const int n0 = w * 16;
  const int rowA = L & 15;
  const int kOff = (L < 16) ? 0 : 2;   // A/B K-pair select per half-wave
  const int colB = L & 15;

  const float bq0 = bq[n0 + colB];
  const float bk0 = bk[n0 + colB];
  const float bv0 = bvo[n0 + colB];
  v8f cq, ck, cv;
  for (int j = 0; j < 8; ++j) { cq[j] = bq0; ck[j] = bk0; cv[j] = bv0; }

  for (int kk = 0; kk < IN_DIM; kk += 4) {
    v2f a;
    a.x = As[rowA][kk + kOff];
    a.y = As[rowA][kk + kOff + 1];
    v2f b;
    b.x = Wq[(kk + kOff) * HD + n0 + colB];
    b.y = Wq[(kk + kOff + 1) * HD + n0 + colB];
    cq = wmma_f32(a, b, cq);
    b.x = Wk[(kk + kOff) * HD + n0 + colB];
    b.y = Wk[(kk + kOff + 1) * HD + n0 + colB];
    ck = wmma_f32(a, b, ck);
    b.x = Wvo[(kk + kOff) * HD + n0 + colB];
    b.y = Wvo[(kk + kOff + 1) * HD + n0 + colB];
    cv = wmma_f32(a, b, cv);
  }

  for (int j = 0; j < 8; ++j) {
    int r = (L < 16) ? j : (j + 8);
    int gr = m0 + r;
    if (gr < N) {
      size_t o = (size_t)gr * HD + n0 + colB;
      Q[o] = cq[j]; Km[o] = ck[j]; Vo[o] = cv[j];
    }
  }
}

// ---------------------------------------------------------------------------
// Edge self-dot: sE[e,h] = || edge_attr[e] @ We[:, h*32:(h+1)*32] + be_h ||^2
// Block = 128 threads = 4 waves = 4 heads, 16 edges per block. Ef tile lives
// only in WMMA accumulators; squares reduced across the 16-lane half-waves.
// ---------------------------------------------------------------------------
__global__ __launch_bounds__(128) void edge_selfdot(
    const float* __restrict__ ea, const float* __restrict__ We,
    const float* __restrict__ be, float* __restrict__ sE, int NE) {
  __shared__ float Ae[16][E_DIM];
  const int tid = threadIdx.x;
  const int e0 = blockIdx.x * 16;

  for (int i = 0; i < 4; ++i) {
    int lin = tid + i * 128;
    int r = lin >> 5, c = lin & 31;
    int ge = e0 + r; if (ge >= NE) ge = NE - 1;
    Ae[r][c] = ea[(size_t)ge * E_DIM + c];
  }
  __syncthreads();

  const int h = tid >> 5, L = tid & 31;
  const int n0 = h * DHEAD;
  const int rowA = L & 15;
  const int kOff = (L < 16) ? 0 : 2;
  const int colB = L & 15;

  v8f c0 = {}, c1 = {};
  for (int kk = 0; kk < E_DIM; kk += 4) {
    v2f a;
    a.x = Ae[rowA][kk + kOff];
    a.y = Ae[rowA][kk + kOff + 1];
    v2f b0, b1;
    b0.x = We[(kk + kOff) * HD + n0 + colB];
    b0.y = We[(kk + kOff + 1) * HD + n0 + colB];
    b1.x = We[(kk + kOff) * HD + n0 + 16 + colB];
    b1.y = We[(kk + kOff + 1) * HD + n0 + 16 + colB];
    c0 = wmma_f32(a, b0, c0);
    c1 = wmma_f32(a, b1, c1);
  }

  const float be0 = be[n0 + colB], be1 = be[n0 + 16 + colB];
  float t[8];
  for (int j = 0; j < 8; ++j) {
    float u = c0[j] + be0, v = c1[j] + be1;
    t[j] = u * u + v * v;
  }
  // reduce over the 16 lanes of each half-wave (cols) -> per-row sum of 32 cols
  for (int m = 1; m <= 8; m <<= 1)
    for (int j = 0; j < 8; ++j) t[j] += __shfl_xor(t[j], m);

  if (L == 0 || L == 16) {
    int rb = (L == 0) ? 0 : 8;
    for (int j = 0; j < 8; ++j) {
      int ge = e0 + rb + j;
      if (ge < NE) sE[(size_t)ge * NHEAD + h] = t[j];
    }
  }
}

// ---------------------------------------------------------------------------
// Edge pass: wave per edge. Gather Q[src],K[dst] (L2-resident), per-head dots,
// softmax over 4 heads, out = sum_h alpha_h * Vo[dst,h,:] + bo, atomic scatter.
// ---------------------------------------------------------------------------
__global__ __launch_bounds__(256) void edge_pass(
    const int* __restrict__ ei, const float* __restrict__ Q,
    const float* __restrict__ Km, const float* __restrict__ Vo,
    const float* __restrict__ sE, const float* __restrict__ bo,
    float* __restrict__ sums, float* __restrict__ cnt, int NE) {
  const int e = blockIdx.x * 8 + (threadIdx.x >> 5);
  if (e >= NE) return;  // wave-uniform
  const int L = threadIdx.x & 31;
  const int src = ei[e];
  const int dst = ei[NE + e];

  // lane -> (head = L/8, 4 dims); dot over D=32 per head
  const int h = L >> 3, d4 = (L & 7) << 2;
  const float4 q = *(const float4*)(Q + (size_t)src * HD + h * DHEAD + d4);
  const float4 k = *(const float4*)(Km + (size_t)dst * HD + h * DHEAD + d4);
  float p = q.x * k.x + q.y * k.y + q.z * k.z + q.w * k.w;
  p += __shfl_xor(p, 1);
  p += __shfl_xor(p, 2);
  p += __shfl_xor(p, 4);

  float sc[NHEAD];
  for (int h2 = 0; h2 < NHEAD; ++h2)
    sc[h2] = (__shfl(p, h2 * 8) + sE[(size_t)e * NHEAD + h2]) * INV_SQRT_D;

  float m = sc[0];
  for (int h2 = 1; h2 < NHEAD; ++h2) m = fmaxf(m, sc[h2]);
  float ex[NHEAD], den = 0.0f;
  for (int h2 = 0; h2 < NHEAD; ++h2) { ex[h2] = __expf(sc[h2] - m); den += ex[h2]; }
  const float inv = 1.0f / den;

  const int j = L;  // 32 output columns
  float acc = bo[j];
  for (int h2 = 0; h2 < NHEAD; ++h2)
    acc += (ex[h2] * inv) * Vo[(size_t)dst * HD + h2 * DHEAD + j];

  atomicAdd(&sums[(size_t)src * OUT_DIM + j], acc);
  if (L == 0) atomicAdd(&cnt[src], 1.0f);
}

// ---------------------------------------------------------------------------
// Finalize: out = sums / max(cnt, 1)
// ---------------------------------------------------------------------------
__global__ void finalize(const float* __restrict__ sums,
                         const float* __restrict__ cnt,
                         float* __restrict__ out, int N) {
  int i = blockIdx.x * blockDim.x + threadIdx.x;
  if (i < N * OUT_DIM)
    out[i] = sums[i] / fmaxf(cnt[i >> 5], 1.0f);
}

extern "C" void kernel_launch(void* const* d_in, const int* in_sizes, int n_in,
                              void* d_out, int out_size, void* d_ws, size_t ws_size,
                              hipStream_t stream) {
  const float* x  = (const float*)d_in[0];
  const int*   ei = (const int*)d_in[1];
  const float* ea = (const float*)d_in[2];
  const float* Wq = (const float*)d_in[3];
  const float* bq = (const float*)d_in[4];
  const float* Wk = (const float*)d_in[5];
  const float* bk = (const float*)d_in[6];
  const float* Wv = (const float*)d_in[7];
  const float* bv = (const float*)d_in[8];
  const float* We = (const float*)d_in[9];
  const float* be = (const float*)d_in[10];
  const float* Wo = (const float*)d_in[11];
  const float* bo = (const float*)d_in[12];

  const int N  = in_sizes[0] / IN_DIM;
  const int NE = in_sizes[1] / 2;

  // workspace layout (floats)
  float* ws   = (float*)d_ws;
  float* Q    = ws;
  float* Km   = Q   + (size_t)N * HD;
  float* Vo   = Km  + (size_t)N * HD;
  float* Wvo  = Vo  + (size_t)N * HD;
  float* bvo  = Wvo + IN_DIM * HD;
  float* sE   = bvo + HD;
  float* sums = sE  + (size_t)NE * NHEAD;
  float* cnt  = sums + (size_t)N * OUT_DIM;

  // 1) zero scatter accumulators (sums ++ cnt contiguous)
  {
    int n = N * OUT_DIM + N;
    zero_kernel<<<(n + 255) / 256, 256, 0, stream>>>(sums, n);
  }
  // 2) fold Wo into Wv
  {
    int n = IN_DIM * HD + HD;
    precompute_wvo<<<(n + 127) / 128, 128, 0, stream>>>(Wv, bv, Wo, Wvo, bvo);
  }
  // 3) fused node GEMM (WMMA): Q, K, Vo
  node_gemm<<<(N + 15) / 16, 256, 0, stream>>>(x, Wq, bq, Wk, bk, Wvo, bvo,
                                               Q, Km, Vo, N);
  // 4) edge self-dot scores (WMMA)
  edge_selfdot<<<(NE + 15) / 16, 128, 0, stream>>>(ea, We, be, sE, NE);
  // 5) edge attention + scatter
  edge_pass<<<(NE + 7) / 8, 256, 0, stream>>>(ei, Q, Km, Vo, sE, bo,
                                              sums, cnt, NE);
  // 6) mean
  {
    int n = N * OUT_DIM;
    finalize<<<(n + 255) / 256, 256, 0, stream>>>(sums, cnt, (float*)d_out, N);
  }
}